// MultiscaleTopKSparseAttention_74741020885071
// MI455X (gfx1250) — compile-verified
//
#include <hip/hip_runtime.h>
#include <hip/hip_bf16.h>
#include <stdint.h>

// Multiscale top-k sparse attention blend, fp32, C = 512.
// One 256-thread workgroup per row. Row staged via ASYNCcnt-tracked
// global->LDS async load. Hybrid bitonic sort: passes with partner
// distance j<=32 run in registers via wave32 shfl_xor + min/max (no
// barriers); only the 6 passes with j>=64 touch LDS. ~15 workgroup
// barriers per row instead of ~47.
//
// Roofline: 64 MiB in + 64 MiB out = 128 MiB -> ~5.8 us at 23.3 TB/s;
// compute is O(1e9) simple VALU ops chip-wide -> memory-bound. The
// barrier-thinned sort keeps sync overhead under the HBM floor.

#define C_DIM    512
#define NTHREADS 256

typedef int v4i_t __attribute__((ext_vector_type(4)));
typedef __attribute__((address_space(1))) v4i_t* g_v4i_ptr;
typedef __attribute__((address_space(3))) v4i_t* l_v4i_ptr;

#if defined(__gfx1250__) && __has_builtin(__builtin_amdgcn_global_load_async_to_lds_b128)
#define USE_ASYNC_LDS 1
#else
#define USE_ASYNC_LDS 0
#endif

// Bitonic compare-exchange for the two elements (2t, 2t+1) held in registers.
// Valid for partner distance j <= 32 (partner is in the same wave).
__device__ __forceinline__ void reg_pass(float& v0, float& v1, int t, int k, int j) {
    const bool desc = (t & (k >> 1)) == 0;     // direction of this k-block
    if (j == 1) {                              // partner is the other register
        const float mx = fmaxf(v0, v1);
        const float mn = fminf(v0, v1);
        v0 = desc ? mx : mn;
        v1 = desc ? mn : mx;
    } else {                                   // partner lane = t ^ (j/2)
        const int  lm      = j >> 1;
        const bool keepmax = (((t & lm) == 0) == desc);
        const float o0 = __shfl_xor(v0, lm, 32);
        const float o1 = __shfl_xor(v1, lm, 32);
        v0 = keepmax ? fmaxf(v0, o0) : fminf(v0, o0);
        v1 = keepmax ? fmaxf(v1, o1) : fminf(v1, o1);
    }
}

// Bitonic compare-exchange through LDS for partner distance j >= 64.
__device__ __forceinline__ void lds_pass(float* s, float& v0, float& v1,
                                         int t, int k, int j) {
    s[2 * t]     = v0;
    s[2 * t + 1] = v1;
    __syncthreads();
    const bool desc    = (t & (k >> 1)) == 0;
    const bool keepmax = (((t & (j >> 1)) == 0) == desc);
    const float o0 = s[(2 * t)     ^ j];
    const float o1 = s[(2 * t + 1) ^ j];
    v0 = keepmax ? fmaxf(v0, o0) : fminf(v0, o0);
    v1 = keepmax ? fmaxf(v1, o1) : fminf(v1, o1);
    __syncthreads();                           // reads done before next overwrite
}

__global__ __launch_bounds__(NTHREADS)
void multiscale_topk_softmax_kernel(const float* __restrict__ attn,
                                    const float* __restrict__ w1,
                                    const float* __restrict__ w2,
                                    const float* __restrict__ w3,
                                    const float* __restrict__ w4,
                                    float* __restrict__ out)
{
    __shared__ float s_vals[C_DIM];   // original order (for output pass)
    __shared__ float s_sort[C_DIM];   // sort scratch / final sorted row
    __shared__ float s_red[8][4];     // per-wave partial sums
    __shared__ float s_S[4];          // softmax denominators for the 4 k's

    const int t = threadIdx.x;
    const long long row = blockIdx.x;
    const float* rowp = attn + row * (long long)C_DIM;

    // ---------- load row into LDS ----------
#if USE_ASYNC_LDS
    if (t < C_DIM / 4) {
        // 128 lanes x 16B = 2 KB row, ASYNCcnt-tracked, no VGPR staging.
        __builtin_amdgcn_global_load_async_to_lds_b128(
            (g_v4i_ptr)(rowp + t * 4),
            (l_v4i_ptr)(&s_vals[t * 4]),
            /*imm offset*/0, /*cpol*/0);
    }
#if __has_builtin(__builtin_amdgcn_s_wait_asynccnt)
    __builtin_amdgcn_s_wait_asynccnt(0);
#else
    asm volatile("s_wait_asynccnt 0" ::: "memory");
#endif
#else
    s_vals[t]       = rowp[t];
    s_vals[t + 256] = rowp[t + 256];
#endif
    __syncthreads();

    // ---------- hybrid bitonic sort, descending ----------
    float v0 = s_vals[2 * t];
    float v1 = s_vals[2 * t + 1];

    // k = 2..64: partner distance <= 32 -> pure register/shuffle passes
    for (int k = 2; k <= 64; k <<= 1)
        for (int j = k >> 1; j >= 1; j >>= 1)
            reg_pass(v0, v1, t, k, j);

    // k = 128, 256, 512: j >= 64 through LDS, j <= 32 in registers
    for (int k = 128; k <= 512; k <<= 1) {
        for (int j = k >> 1; j >= 64; j >>= 1)
            lds_pass(s_sort, v0, v1, t, k, j);
        for (int j = 32; j >= 1; j >>= 1)
            reg_pass(v0, v1, t, k, j);
    }

    // publish sorted row (descending): regs hold sorted[2t], sorted[2t+1]
    s_sort[2 * t]     = v0;
    s_sort[2 * t + 1] = v1;
    __syncthreads();

    // thresholds: k-th largest value, k in {256, 341, 384, 409}
    const float m  = s_sort[0];
    const float T0 = s_sort[255];
    const float T1 = s_sort[340];
    const float T2 = s_sort[383];
    const float T3 = s_sort[408];

    // ---------- partial softmax denominators S_k = sum_{i<k} exp(sorted[i]-m) ----------
    const int   i0 = 2 * t;
    const int   i1 = 2 * t + 1;
    const float e0 = __expf(v0 - m);
    const float e1 = __expf(v1 - m);

    float l[4];
    l[0] = ((i0 < 256) ? e0 : 0.0f) + ((i1 < 256) ? e1 : 0.0f);
    l[1] = ((i0 < 341) ? e0 : 0.0f) + ((i1 < 341) ? e1 : 0.0f);
    l[2] = ((i0 < 384) ? e0 : 0.0f) + ((i1 < 384) ? e1 : 0.0f);
    l[3] = ((i0 < 409) ? e0 : 0.0f) + ((i1 < 409) ? e1 : 0.0f);

#pragma unroll
    for (int off = 16; off > 0; off >>= 1) {
#pragma unroll
        for (int j = 0; j < 4; ++j)
            l[j] += __shfl_xor(l[j], off, 32);   // wave32 butterfly
    }
    const int wave = t >> 5;
    const int lane = t & 31;
    if (lane == 0) {
#pragma unroll
        for (int j = 0; j < 4; ++j) s_red[wave][j] = l[j];
    }
    __syncthreads();
    if (t < 4) {
        float s = 0.0f;
#pragma unroll
        for (int w = 0; w < 8; ++w) s += s_red[w][t];
        s_S[t] = s;
    }
    __syncthreads();

    // blend coefficients (uniform across the workgroup)
    const float c0 = w1[0] / s_S[0];
    const float c1 = w2[0] / s_S[1];
    const float c2 = w3[0] / s_S[2];
    const float c3 = w4[0] / s_S[3];

    // ---------- output: exp(v-m) * sum of coefficients whose threshold v clears ----------
    float* orow = out + row * (long long)C_DIM;
#pragma unroll
    for (int half = 0; half < 2; ++half) {
        const int i = t + half * 256;
        const float v = s_vals[i];
        float f = 0.0f;
        if (v >= T0) f += c0;
        if (v >= T1) f += c1;
        if (v >= T2) f += c2;
        if (v >= T3) f += c3;
        orow[i] = __expf(v - m) * f;   // f == 0 below T3 -> exact zero
    }
}

extern "C" void kernel_launch(void* const* d_in, const int* in_sizes, int n_in,
                              void* d_out, int out_size, void* d_ws, size_t ws_size,
                              hipStream_t stream) {
    (void)n_in; (void)out_size; (void)d_ws; (void)ws_size;
    const float* attn = (const float*)d_in[0];
    const float* w1   = (const float*)d_in[1];
    const float* w2   = (const float*)d_in[2];
    const float* w3   = (const float*)d_in[3];
    const float* w4   = (const float*)d_in[4];
    float* out = (float*)d_out;

    const int rows = in_sizes[0] / C_DIM;   // b * nh * C = 32768
    multiscale_topk_softmax_kernel<<<rows, NTHREADS, 0, stream>>>(
        attn, w1, w2, w3, w4, out);
}